// HierarchicalClassifier_61435212202267
// MI455X (gfx1250) — compile-verified
//
#include <hip/hip_runtime.h>
#include <math.h>

typedef __attribute__((ext_vector_type(2))) float v2f;
typedef __attribute__((ext_vector_type(8))) float v8f;

#define N_DELTA 4
#define N_THETA 8
#define N_GAMMA 32
#define N_TOT   44
#define NPAD    48      // 44 padded to 3 x 16 WMMA tiles
#define N_CLS   5
#define KDIM    128
#define NCHUNK  (KDIM/4) // 32 K-chunks of 4
#define ROWS_PER_BLOCK 128
#define THREADS 256
#define DT_        0.01f
#define COUPLING_  2.0f
#define PAC_DEPTH_ 0.3f
#define TWO_PI_    6.28318530717958647692f

#if __has_builtin(__builtin_amdgcn_tanhf)
#define fast_tanh(v) __builtin_amdgcn_tanhf(v)
#else
#define fast_tanh(v) tanhf(v)
#endif

__global__ __launch_bounds__(THREADS)
void hier_classifier_kernel(const float* __restrict__ x,
                            const float* __restrict__ Wd,
                            const float* __restrict__ Wt,
                            const float* __restrict__ Wg,
                            const float* __restrict__ Wc,
                            const float* __restrict__ bc,
                            float* __restrict__ out)
{
    // W stored swizzled into B-fragment order: 64-bit entry (pair) index
    //   p(c,g,n) = (c*2 + g)*NPAD + n
    // holds (W[4c+2g][n], W[4c+2g+1][n]) so each B fragment is one ds_load_b64
    // straight into a WMMA-ready even-aligned VGPR pair.
    __shared__ float Wswz[NCHUNK * 2 * NPAD * 2];   // 24576 B; reused as h-tile later
    __shared__ float Wcp[NPAD * N_CLS];             // classifier weights (gamma rows get *C)
    __shared__ float bcs[N_CLS];
    __shared__ float Cs;                            // scalar PAC/damping factor

    const int tid = threadIdx.x;

    // ---- stage fused W = [Wd | Wt | Wg] (128 x 48, zero-padded), swizzled ----
    for (int i = tid; i < KDIM * NPAD; i += THREADS) {
        const int k = i / NPAD;
        const int n = i - k * NPAD;
        float v = 0.0f;
        if (n < N_DELTA)                 v = Wd[k * N_DELTA + n];
        else if (n < N_DELTA + N_THETA)  v = Wt[k * N_THETA + (n - N_DELTA)];
        else if (n < N_TOT)              v = Wg[k * N_GAMMA + (n - N_DELTA - N_THETA)];
        const int c  = k >> 2;
        const int g  = (k >> 1) & 1;
        const int lo = k & 1;
        Wswz[2 * ((c * 2 + g) * NPAD + n) + lo] = v;
    }
    for (int i = tid; i < NPAD * N_CLS; i += THREADS)
        Wcp[i] = (i < N_TOT * N_CLS) ? Wc[i] : 0.0f;
    if (tid < N_CLS) bcs[tid] = bc[tid];

    // ---- thread 0: 5-step phase recurrence -> single scalar C ----
    // Phases are batch-independent; a_d/a_t are fixed points of the damped
    // update (a == drive), a_g collapses to drive * C.
    if (tid == 0) {
        float w_d[N_DELTA], w_t[N_THETA], ph_t[N_THETA];
        for (int i = 0; i < N_DELTA; ++i) w_d[i] = TWO_PI_ * (1.0f + 3.0f * i / (float)(N_DELTA - 1));
        for (int j = 0; j < N_THETA; ++j) w_t[j] = TWO_PI_ * (4.0f + 4.0f * j / (float)(N_THETA - 1));
        for (int j = 0; j < N_THETA; ++j) ph_t[j] = 0.0f;
        float C = 1.0f;
        for (int s = 0; s < 5; ++s) {
            float ssd = 0.0f, ccd = 0.0f;
            for (int i = 0; i < N_DELTA; ++i) {
                const float p = (float)s * DT_ * w_d[i];
                ssd += sinf(p); ccd += cosf(p);
            }
            const float mp_d = atan2f(ssd * (1.0f / N_DELTA), ccd * (1.0f / N_DELTA));
            float sst = 0.0f, cct = 0.0f;
            for (int j = 0; j < N_THETA; ++j) { sst += sinf(ph_t[j]); cct += cosf(ph_t[j]); }
            const float mp_t = atan2f(sst * (1.0f / N_THETA), cct * (1.0f / N_THETA));
            const float pac = 1.0f + PAC_DEPTH_ * cosf(mp_t);
            C = (1.0f - DT_) * C + DT_ * pac;
            for (int j = 0; j < N_THETA; ++j)
                ph_t[j] += DT_ * (w_t[j] + COUPLING_ * sinf(mp_d - ph_t[j]));
        }
        Cs = C;
    }
    __syncthreads();

    // ---- fold gamma scale C into classifier weight rows 12..43 ----
    {
        const float C = Cs;
        for (int i = tid; i < N_GAMMA * N_CLS; i += THREADS)
            Wcp[(N_DELTA + N_THETA) * N_CLS + i] *= C;
        // consumed only after the next __syncthreads()
    }

    // ---- WMMA GEMM: 16-row tile per wave, 3 N-tiles, K=128 in chunks of 4 ----
    const int wave  = tid >> 5;
    const int lane  = tid & 31;
    const int mrow  = lane & 15;        // matrix row (A) / column (B) within tile
    const int g     = lane >> 4;        // 0: K+0/K+1, 1: K+2/K+3
    const size_t row0 = (size_t)blockIdx.x * ROWS_PER_BLOCK + (size_t)wave * 16;
    const float* xp = x + (row0 + (size_t)mrow) * KDIM + 2 * g;
    const v2f* __restrict__ Bp = (const v2f*)Wswz;

    v8f c0 = {}; v8f c1 = {}; v8f c2 = {};
    #pragma unroll 8
    for (int c = 0; c < NCHUNK; ++c) {
        // A fragment (16x4 f32): lane holds x[row][4c+2g], x[row][4c+2g+1]
        const v2f a = *(const v2f*)(xp + 4 * c);
        // B fragments (4x16 f32): one b64 each, already in fragment order
        const int bb = (c * 2 + g) * NPAD + mrow;
        const v2f b0 = Bp[bb];
        const v2f b1 = Bp[bb + 16];
        const v2f b2 = Bp[bb + 32];
        c0 = __builtin_amdgcn_wmma_f32_16x16x4_f32(false, a, false, b0, (short)0, c0, false, false);
        c1 = __builtin_amdgcn_wmma_f32_16x16x4_f32(false, a, false, b1, (short)0, c1, false, false);
        c2 = __builtin_amdgcn_wmma_f32_16x16x4_f32(false, a, false, b2, (short)0, c2, false, false);
    }

    // ---- tanh + stash activations into LDS (reuse W region; GEMM is done) ----
    __syncthreads();
    float* h = Wswz;
    const int lrbase = wave * 16 + g * 8; // C layout: vgpr i -> row i (+8 for hi lanes)
    #pragma unroll
    for (int i = 0; i < 8; ++i) {
        h[(lrbase + i) * NPAD + mrow     ] = fast_tanh(c0[i]);
        h[(lrbase + i) * NPAD + mrow + 16] = fast_tanh(c1[i]);
        h[(lrbase + i) * NPAD + mrow + 32] = fast_tanh(c2[i]);
    }
    __syncthreads();

    // ---- tiny 44x5 projection: one thread per row ----
    if (tid < ROWS_PER_BLOCK) {
        const float* hr = &h[tid * NPAD];
        float acc[N_CLS];
        #pragma unroll
        for (int c = 0; c < N_CLS; ++c) acc[c] = bcs[c];
        #pragma unroll 4
        for (int n = 0; n < N_TOT; ++n) {
            const float hv = hr[n];
            #pragma unroll
            for (int c = 0; c < N_CLS; ++c) acc[c] += hv * Wcp[n * N_CLS + c];
        }
        const size_t orow = (size_t)blockIdx.x * ROWS_PER_BLOCK + (size_t)tid;
        #pragma unroll
        for (int c = 0; c < N_CLS; ++c) out[orow * N_CLS + c] = acc[c];
    }
}

extern "C" void kernel_launch(void* const* d_in, const int* in_sizes, int n_in,
                              void* d_out, int out_size, void* d_ws, size_t ws_size,
                              hipStream_t stream) {
    const float* x  = (const float*)d_in[0];
    const float* Wd = (const float*)d_in[1];
    const float* Wt = (const float*)d_in[2];
    const float* Wg = (const float*)d_in[3];
    const float* Wc = (const float*)d_in[4];
    const float* bc = (const float*)d_in[5];
    float* out = (float*)d_out;

    const int Brows = in_sizes[0] / KDIM;                       // 524288
    const int grid  = (Brows + ROWS_PER_BLOCK - 1) / ROWS_PER_BLOCK;
    hier_classifier_kernel<<<grid, THREADS, 0, stream>>>(x, Wd, Wt, Wg, Wc, bc, out);
}